// DynamicRouting_86766929314212
// MI455X (gfx1250) — compile-verified
//
#include <hip/hip_runtime.h>
#include <hip/hip_bf16.h>

typedef __attribute__((ext_vector_type(2))) float v2f;
typedef __attribute__((ext_vector_type(8))) float v8f;

namespace {
constexpr int Bc = 8;     // batch
constexpr int Ic = 64;    // in capsules
constexpr int Kc = 16;    // in dim
constexpr int Lc = 256;   // seq len
constexpr int Oc = 16;    // out capsules
constexpr int Dc = 32;    // out dim
constexpr int LT = 16;    // l-tile (WMMA N)
constexpr int ITERS = 3;
constexpr int XSS = Kc + 2;           // padded k-stride for xs[i][n][k]
constexpr int NTHREADS = Oc * 32;     // one wave32 per out-capsule
constexpr int LDS_FLOATS = Ic * LT * XSS   // xs
                         + Oc * Ic * LT    // routing logits
                         + Ic * LT         // softmax max
                         + Ic * LT;        // softmax 1/denominator
}

// Compute the two 16x16 u_hat tiles (d-halves) for one (o,i):
//   u = W[o,i] (32x16) @ xtile (16x16), via 4 chained V_WMMA_F32_16X16X4_F32.
// A-frag fp32 16x4 layout: lane(m,h) holds A[m, 4c+2h .. 4c+2h+1]  (contiguous b64)
// B-frag fp32 4x16 layout: lane(n,h) holds B[4c+2h .. 4c+2h+1, n]  (contiguous b64 in xs[i][n][k])
__device__ __forceinline__ void wmma_u(const float* __restrict__ Wp,
                                       const float* __restrict__ xrow,
                                       int arow, v8f& u0, v8f& u1)
{
#pragma unroll
    for (int c = 0; c < 4; ++c) {
        v2f bf = *(const v2f*)(xrow + 4 * c);
        v2f a0 = *(const v2f*)(Wp + arow + 4 * c);
        v2f a1 = *(const v2f*)(Wp + arow + 16 * Kc + 4 * c);
        u0 = __builtin_amdgcn_wmma_f32_16x16x4_f32(false, a0, false, bf, (short)0, u0, false, false);
        u1 = __builtin_amdgcn_wmma_f32_16x16x4_f32(false, a1, false, bf, (short)0, u1, false, false);
    }
}

__global__ __launch_bounds__(NTHREADS)
void DynamicRouting_86766929314212_kernel(const float* __restrict__ x,
                                          const float* __restrict__ W,
                                          float* __restrict__ out)
{
    extern __shared__ float lds[];
    float* xs = lds;                        // [Ic][LT][XSS]  x-tile, transposed
    float* bl = xs + Ic * LT * XSS;         // [Oc][Ic][LT]   routing logits
    float* mx = bl + Oc * Ic * LT;          // [Ic][LT]       softmax max over o
    float* rd = mx + Ic * LT;               // [Ic][LT]       1/sum(exp)

    const int tid  = threadIdx.x;
    const int lane = tid & 31;
    const int o    = tid >> 5;              // wave id == out capsule
    const int n    = lane & 15;             // tile column (= l offset) / A row m
    const int h    = lane >> 4;             // lane half
    const int arow = n * Kc + 2 * h;        // per-lane W row offset for A-frags

    const int tilesPerB = Lc / LT;
    const int b  = blockIdx.x / tilesPerB;
    const int l0 = (blockIdx.x % tilesPerB) * LT;

    const float* Wo = W + (size_t)o * Ic * Dc * Kc;

    // ---- stage x[b, :, :, l0:l0+16] into LDS, transposed to xs[i][n][k] ----
    for (int idx = tid; idx < Ic * Kc * LT; idx += NTHREADS) {
        int i  = idx >> 8;        // / (Kc*LT)
        int r  = idx & 255;
        int k  = r >> 4;
        int nn = r & 15;
        xs[(i * LT + nn) * XSS + k] = x[((size_t)(b * Ic + i) * Kc + k) * Lc + l0 + nn];
    }
    // ---- zero routing logits ----
    for (int idx = tid; idx < Oc * Ic * LT; idx += NTHREADS) bl[idx] = 0.0f;

    v8f v0, v1;   // final squashed output tiles live here after last iteration

    for (int it = 0; it < ITERS; ++it) {
        // ---- softmax stats over o (couples waves) ----
        __syncthreads();
        for (int idx = tid; idx < Ic * LT; idx += NTHREADS) {
            float m = bl[idx];
            for (int oo = 1; oo < Oc; ++oo) m = fmaxf(m, bl[oo * Ic * LT + idx]);
            float den = 0.0f;
            for (int oo = 0; oo < Oc; ++oo) den += __expf(bl[oo * Ic * LT + idx] - m);
            mx[idx] = m;
            rd[idx] = 1.0f / den;
        }
        __syncthreads();

        // ---- pass A: s[o,n,d] = sum_i c[o,i,n] * u_hat[o,i,n,d]  (u via WMMA) ----
        v8f s0 = {0.f,0.f,0.f,0.f,0.f,0.f,0.f,0.f};
        v8f s1 = {0.f,0.f,0.f,0.f,0.f,0.f,0.f,0.f};
        for (int i = 0; i < Ic; ++i) {
            v8f u0 = {0.f,0.f,0.f,0.f,0.f,0.f,0.f,0.f};
            v8f u1 = {0.f,0.f,0.f,0.f,0.f,0.f,0.f,0.f};
            wmma_u(Wo + i * Dc * Kc, xs + (i * LT + n) * XSS + 2 * h, arow, u0, u1);
            float cb = __expf(bl[(o * Ic + i) * LT + n] - mx[i * LT + n]) * rd[i * LT + n];
#pragma unroll
            for (int j = 0; j < 8; ++j) { s0[j] += cb * u0[j]; s1[j] += cb * u1[j]; }
        }

        // ---- squash over d: lane-local + cross-half shuffle ----
        float sq = 0.0f;
#pragma unroll
        for (int j = 0; j < 8; ++j) sq += s0[j] * s0[j] + s1[j] * s1[j];
        sq += __shfl_xor(sq, 16, 32);
        const float scale = (sq / (1.0f + sq)) * rsqrtf(sq + 1e-8f);
#pragma unroll
        for (int j = 0; j < 8; ++j) { s0[j] *= scale; s1[j] *= scale; }
        v0 = s0; v1 = s1;

        if (it + 1 < ITERS) {
            // ---- pass B: b[o,i,n] += sum_d u_hat[o,i,n,d] * v[o,n,d]  (recompute u) ----
            for (int i = 0; i < Ic; ++i) {
                v8f u0 = {0.f,0.f,0.f,0.f,0.f,0.f,0.f,0.f};
                v8f u1 = {0.f,0.f,0.f,0.f,0.f,0.f,0.f,0.f};
                wmma_u(Wo + i * Dc * Kc, xs + (i * LT + n) * XSS + 2 * h, arow, u0, u1);
                float dot = 0.0f;
#pragma unroll
                for (int j = 0; j < 8; ++j) dot += u0[j] * v0[j] + u1[j] * v1[j];
                dot += __shfl_xor(dot, 16, 32);
                if (h == 0) bl[(o * Ic + i) * LT + n] += dot;
            }
        }
    }

    // ---- write v as [B,O,D,L]: D-tile element (j,h,t) -> d = 16t + j + 8h, col n -> l0+n ----
    const size_t obase = ((size_t)(b * Oc + o) * Dc) * Lc + l0 + n;
#pragma unroll
    for (int j = 0; j < 8; ++j) {
        int d = j + 8 * h;
        out[obase + (size_t)d * Lc]        = v0[j];
        out[obase + (size_t)(d + 16) * Lc] = v1[j];
    }
}

extern "C" void kernel_launch(void* const* d_in, const int* in_sizes, int n_in,
                              void* d_out, int out_size, void* d_ws, size_t ws_size,
                              hipStream_t stream)
{
    (void)in_sizes; (void)n_in; (void)out_size; (void)d_ws; (void)ws_size;
    const float* x = (const float*)d_in[0];   // [B, I, K, L]
    const float* W = (const float*)d_in[1];   // [O, I, D, K]
    float* out = (float*)d_out;               // [B, O, D, L]

    constexpr size_t shmem = (size_t)LDS_FLOATS * sizeof(float);  // 147456 B
    dim3 grid(Bc * (Lc / LT));                                    // 128 workgroups
    DynamicRouting_86766929314212_kernel<<<grid, NTHREADS, shmem, stream>>>(x, W, out);
}